// MambaBlock_40896678592878
// MI455X (gfx1250) — compile-verified
//
#include <hip/hip_runtime.h>
#include <math.h>

// ---------------------------------------------------------------------------
// Mamba block for MI455X (gfx1250, wave32, WMMA)
// B=2, L=2048, D_MODEL=1024, D_INNER=2048, D_STATE=16, D_CONV=4, DT_RANK=64
// ---------------------------------------------------------------------------

typedef float v2f __attribute__((ext_vector_type(2)));
typedef float v8f __attribute__((ext_vector_type(8)));
typedef int   v4i __attribute__((ext_vector_type(4)));

#define AS1 __attribute__((address_space(1)))
#define AS3 __attribute__((address_space(3)))

#if __has_builtin(__builtin_amdgcn_global_load_async_to_lds_b128) && \
    __has_builtin(__builtin_amdgcn_s_wait_asynccnt)
#define USE_ASYNC_LDS 1
#else
#define USE_ASYNC_LDS 0
#endif

#define BATCH     2
#define SEQ       2048
#define DMODEL    1024
#define DINNER    2048
#define DSTATE    16
#define DCONV     4
#define DTRANK    64
#define ROWS      (BATCH * SEQ)          // 4096
#define XDBL_W    (DTRANK + 2 * DSTATE)  // 96

#define LDS_STRIDE 36   // padded row stride (floats): 144B, 16B aligned

// ---------------------------------------------------------------------------
// 1) LayerNorm: one block per row, 256 threads, float4 per thread
// ---------------------------------------------------------------------------
__global__ __launch_bounds__(256) void ln_kernel(const float* __restrict__ x,
                                                 const float* __restrict__ g,
                                                 const float* __restrict__ b,
                                                 float* __restrict__ out) {
    const int row = blockIdx.x;
    const int t = threadIdx.x;
    const float4 v = reinterpret_cast<const float4*>(x + (size_t)row * DMODEL)[t];
    float s1 = v.x + v.y + v.z + v.w;
    float s2 = v.x * v.x + v.y * v.y + v.z * v.z + v.w * v.w;

    __shared__ float r1[256];
    __shared__ float r2[256];
    r1[t] = s1; r2[t] = s2;
    __syncthreads();
    for (int off = 128; off > 0; off >>= 1) {
        if (t < off) { r1[t] += r1[t + off]; r2[t] += r2[t + off]; }
        __syncthreads();
    }
    const float mu  = r1[0] * (1.0f / DMODEL);
    const float var = r2[0] * (1.0f / DMODEL) - mu * mu;
    const float rs  = rsqrtf(var + 1e-5f);

    const float4 gv = reinterpret_cast<const float4*>(g)[t];
    const float4 bv = reinterpret_cast<const float4*>(b)[t];
    float4 o;
    o.x = (v.x - mu) * rs * gv.x + bv.x;
    o.y = (v.y - mu) * rs * gv.y + bv.y;
    o.z = (v.z - mu) * rs * gv.z + bv.z;
    o.w = (v.w - mu) * rs * gv.w + bv.w;
    reinterpret_cast<float4*>(out + (size_t)row * DMODEL)[t] = o;
}

// ---------------------------------------------------------------------------
// 2a) BIG fp32 WMMA GEMM: C[M,N] = A[M,K] * W[N,K]^T (+ epilogue)
//     Block: 256 threads = 8 waves (4Mx2N), wave tile 32x32 (2x2 WMMA tiles)
//     -> block tile 128M x 64N.  Requires M%128==0, N%64==0, K%32==0.
//     Double-buffered LDS panels filled by async global->LDS copies.
//     mode 0: plain   mode 1: softplus(acc+bias[n])   mode 2: acc+bias[n]+res
// ---------------------------------------------------------------------------
__device__ __forceinline__ void stage_panels_big(
        const float* __restrict__ A, int lda,
        const float* __restrict__ W, int ldw,
        int gm0, int gn0, int k0,
        float* __restrict__ As, float* __restrict__ Ws, int t) {
    const int ar = t >> 1;            // 0..127
    const int ac = (t & 1) * 16;      // 0 or 16
    const int wr = t >> 2;            // 0..63
    const int wc = (t & 3) * 8;       // 0,8,16,24
    const float* asrc = A + (size_t)(gm0 + ar) * lda + k0 + ac;
    float*       adst = As + ar * LDS_STRIDE + ac;
    const float* wsrc = W + (size_t)(gn0 + wr) * ldw + k0 + wc;
    float*       wdst = Ws + wr * LDS_STRIDE + wc;
#if USE_ASYNC_LDS
#pragma unroll
    for (int q = 0; q < 4; ++q)
        __builtin_amdgcn_global_load_async_to_lds_b128(
            (AS1 v4i*)(asrc + q * 4), (AS3 v4i*)(adst + q * 4), 0, 0);
#pragma unroll
    for (int q = 0; q < 2; ++q)
        __builtin_amdgcn_global_load_async_to_lds_b128(
            (AS1 v4i*)(wsrc + q * 4), (AS3 v4i*)(wdst + q * 4), 0, 0);
#else
#pragma unroll
    for (int q = 0; q < 4; ++q)
        reinterpret_cast<float4*>(adst)[q] =
            reinterpret_cast<const float4*>(asrc)[q];
#pragma unroll
    for (int q = 0; q < 2; ++q)
        reinterpret_cast<float4*>(wdst)[q] =
            reinterpret_cast<const float4*>(wsrc)[q];
#endif
}

__global__ __launch_bounds__(256) void gemm_wmma_big_kernel(
        const float* __restrict__ A, int lda,
        const float* __restrict__ W, int ldw,
        float* __restrict__ C, int ldc,
        int M, int N, int K,
        const float* __restrict__ bias,
        const float* __restrict__ res,
        int mode) {
    __shared__ float As[2][128 * LDS_STRIDE];   // 2 x 18KB
    __shared__ float Ws[2][64 * LDS_STRIDE];    // 2 x 9KB

    const int t    = threadIdx.x;
    const int wave = t >> 5;          // 0..7
    const int lane = t & 31;
    const int wm   = wave >> 1;       // 0..3  (M direction, 32 rows each)
    const int wn   = wave & 1;        // 0..1  (N direction, 32 cols each)

    const int gm0 = blockIdx.y * 128;
    const int gn0 = blockIdx.x * 64;

    const int mlane = lane & 15;
    const int koff  = (lane >> 4) * 2;     // 0 or 2

    v8f acc00 = {}, acc01 = {}, acc10 = {}, acc11 = {};

    const int npanels = K >> 5;            // K/32

    stage_panels_big(A, lda, W, ldw, gm0, gn0, 0, As[0], Ws[0], t);

    for (int p = 0; p < npanels; ++p) {
        const int buf = p & 1;
        if (p + 1 < npanels) {
            stage_panels_big(A, lda, W, ldw, gm0, gn0, (p + 1) * 32,
                             As[(p + 1) & 1], Ws[(p + 1) & 1], t);
#if USE_ASYNC_LDS
            __builtin_amdgcn_s_wait_asynccnt(6);   // drain previous panel only
#endif
        } else {
#if USE_ASYNC_LDS
            __builtin_amdgcn_s_wait_asynccnt(0);
#endif
        }
        __syncthreads();

        const float* a0 = &As[buf][(wm * 32 + mlane) * LDS_STRIDE];
        const float* a1 = a0 + 16 * LDS_STRIDE;
        const float* b0 = &Ws[buf][(wn * 32 + mlane) * LDS_STRIDE];
        const float* b1 = b0 + 16 * LDS_STRIDE;
#pragma unroll
        for (int kk = 0; kk < 32; kk += 4) {
            const float2 a0f = *reinterpret_cast<const float2*>(a0 + kk + koff);
            const float2 a1f = *reinterpret_cast<const float2*>(a1 + kk + koff);
            const float2 b0f = *reinterpret_cast<const float2*>(b0 + kk + koff);
            const float2 b1f = *reinterpret_cast<const float2*>(b1 + kk + koff);
            v2f fa0; fa0.x = a0f.x; fa0.y = a0f.y;
            v2f fa1; fa1.x = a1f.x; fa1.y = a1f.y;
            v2f fb0; fb0.x = b0f.x; fb0.y = b0f.y;
            v2f fb1; fb1.x = b1f.x; fb1.y = b1f.y;
            acc00 = __builtin_amdgcn_wmma_f32_16x16x4_f32(false, fa0, false, fb0, (short)0, acc00, false, false);
            acc01 = __builtin_amdgcn_wmma_f32_16x16x4_f32(false, fa0, false, fb1, (short)0, acc01, false, false);
            acc10 = __builtin_amdgcn_wmma_f32_16x16x4_f32(false, fa1, false, fb0, (short)0, acc10, false, false);
            acc11 = __builtin_amdgcn_wmma_f32_16x16x4_f32(false, fa1, false, fb1, (short)0, acc11, false, false);
        }
        __syncthreads();
    }

    // ---- epilogue: C/D layout -> VGPR v = row v (+8 for lanes 16..31) ----
    const int hi = (lane >> 4) << 3;
#pragma unroll
    for (int i = 0; i < 2; ++i) {
#pragma unroll
        for (int j = 0; j < 2; ++j) {
            const v8f acc = (i == 0) ? (j == 0 ? acc00 : acc01)
                                     : (j == 0 ? acc10 : acc11);
            const int n = gn0 + wn * 32 + j * 16 + mlane;
            const int mbase = gm0 + wm * 32 + i * 16 + hi;
#pragma unroll
            for (int v = 0; v < 8; ++v) {
                const int m = mbase + v;
                float val = acc[v];
                if (mode == 1) {
                    val += bias[n];
                    val = (val > 20.0f) ? val : log1pf(__expf(val));
                } else if (mode == 2) {
                    val += bias[n] + res[(size_t)m * ldc + n];
                }
                C[(size_t)m * ldc + n] = val;
            }
        }
    }
}

// ---------------------------------------------------------------------------
// 2b) SMALL fp32 WMMA GEMM (16x16 per wave, 64Mx32N per block) — for N=96
// ---------------------------------------------------------------------------
__global__ __launch_bounds__(256) void gemm_wmma_kernel(
        const float* __restrict__ A, int lda,
        const float* __restrict__ W, int ldw,
        float* __restrict__ C, int ldc,
        int M, int N, int K,
        const float* __restrict__ bias,
        const float* __restrict__ res,
        int mode) {
    __shared__ float As[64 * LDS_STRIDE];
    __shared__ float Ws[32 * LDS_STRIDE];

    const int t    = threadIdx.x;
    const int wave = t >> 5;
    const int lane = t & 31;
    const int wm   = wave >> 1;
    const int wn   = wave & 1;

    const int gm0 = blockIdx.y * 64;
    const int gn0 = blockIdx.x * 32;

    const int mlane = lane & 15;
    const int koff  = (lane >> 4) * 2;

    const int ar = t >> 2;
    const int ac = (t & 3) * 8;
    const int wr = t >> 3;
    const int wc = (t & 7) * 4;

    v8f acc = {};

    for (int k0 = 0; k0 < K; k0 += 32) {
        {
            const float* src = A + (size_t)(gm0 + ar) * lda + k0 + ac;
            const float4 p0 = reinterpret_cast<const float4*>(src)[0];
            const float4 p1 = reinterpret_cast<const float4*>(src)[1];
            float* dst = &As[ar * LDS_STRIDE + ac];
            reinterpret_cast<float4*>(dst)[0] = p0;
            reinterpret_cast<float4*>(dst)[1] = p1;
        }
        {
            const float* src = W + (size_t)(gn0 + wr) * ldw + k0 + wc;
            *reinterpret_cast<float4*>(&Ws[wr * LDS_STRIDE + wc]) =
                *reinterpret_cast<const float4*>(src);
        }
        if (k0 + 32 < K) {
            __builtin_prefetch(A + (size_t)(gm0 + ar) * lda + k0 + 32 + ac, 0, 3);
            __builtin_prefetch(W + (size_t)(gn0 + wr) * ldw + k0 + 32 + wc, 0, 3);
        }
        __syncthreads();

        const float* arow = &As[(wm * 16 + mlane) * LDS_STRIDE];
        const float* wrow = &Ws[(wn * 16 + mlane) * LDS_STRIDE];
#pragma unroll
        for (int kk = 0; kk < 32; kk += 4) {
            const float2 af = *reinterpret_cast<const float2*>(arow + kk + koff);
            const float2 bf = *reinterpret_cast<const float2*>(wrow + kk + koff);
            v2f a; a.x = af.x; a.y = af.y;
            v2f b; b.x = bf.x; b.y = bf.y;
            acc = __builtin_amdgcn_wmma_f32_16x16x4_f32(
                false, a, false, b, (short)0, acc, false, false);
        }
        __syncthreads();
    }

    const int n = gn0 + wn * 16 + mlane;
    const int mbase = gm0 + wm * 16 + ((lane >> 4) << 3);
#pragma unroll
    for (int v = 0; v < 8; ++v) {
        const int m = mbase + v;
        float val = acc[v];
        if (mode == 1) {
            val += bias[n];
            val = (val > 20.0f) ? val : log1pf(__expf(val));
        } else if (mode == 2) {
            val += bias[n] + res[(size_t)m * ldc + n];
        }
        C[(size_t)m * ldc + n] = val;
    }
}

// ---------------------------------------------------------------------------
// 3) Depthwise causal conv(4) + SiLU, reading u = xz[..., :DINNER]
// ---------------------------------------------------------------------------
__global__ __launch_bounds__(256) void conv_silu_kernel(
        const float* __restrict__ xz,
        const float* __restrict__ cw,
        const float* __restrict__ cb,
        float* __restrict__ ucv) {
    const int idx = blockIdx.x * 256 + threadIdx.x;
    const int d   = idx & (DINNER - 1);
    const int row = idx >> 11;
    const int l   = row & (SEQ - 1);

    float acc = cb[d];
#pragma unroll
    for (int k = 0; k < DCONV; ++k) {
        const int li = l + k - (DCONV - 1);
        if (li >= 0) {
            acc += xz[(size_t)(row + k - (DCONV - 1)) * (2 * DINNER) + d] * cw[d * DCONV + k];
        }
    }
    const float s = acc / (1.0f + __expf(-acc));
    ucv[(size_t)row * DINNER + d] = s;
}

// ---------------------------------------------------------------------------
// 4) Selective scan: one thread per (b, d_inner); fuses D-skip + silu(z) gate
// ---------------------------------------------------------------------------
__global__ __launch_bounds__(256) void scan_kernel(
        const float* __restrict__ dt,
        float* __restrict__ ucv,
        const float* __restrict__ xz,
        const float* __restrict__ xdbl,
        const float* __restrict__ A_log,
        const float* __restrict__ Dp) {
    const int idx = blockIdx.x * blockDim.x + threadIdx.x;
    const int d = idx & (DINNER - 1);
    const int b = idx >> 11;

    float negA[DSTATE];
#pragma unroll
    for (int s = 0; s < DSTATE; ++s) negA[s] = -__expf(A_log[d * DSTATE + s]);
    const float Dd = Dp[d];

    float h[DSTATE];
#pragma unroll
    for (int s = 0; s < DSTATE; ++s) h[s] = 0.0f;

    for (int l = 0; l < SEQ; ++l) {
        const int row = b * SEQ + l;
        const float dtv = dt[(size_t)row * DINNER + d];
        const float uv  = ucv[(size_t)row * DINNER + d];
        const float zv  = xz[(size_t)row * (2 * DINNER) + DINNER + d];

        const float4* bc = reinterpret_cast<const float4*>(
            xdbl + (size_t)row * XDBL_W + DTRANK);
        float4 bcv[8];
#pragma unroll
        for (int q = 0; q < 8; ++q) bcv[q] = bc[q];
        const float* Bv = reinterpret_cast<const float*>(&bcv[0]);
        const float* Cv = Bv + DSTATE;

        const float du = dtv * uv;
        float y = 0.0f;
#pragma unroll
        for (int s = 0; s < DSTATE; ++s) {
            const float dA = __expf(dtv * negA[s]);
            h[s] = dA * h[s] + du * Bv[s];
            y += h[s] * Cv[s];
        }
        y += uv * Dd;
        const float sz = zv / (1.0f + __expf(-zv));
        ucv[(size_t)row * DINNER + d] = y * sz;
    }
}

// ---------------------------------------------------------------------------
// Launcher
// ---------------------------------------------------------------------------
extern "C" void kernel_launch(void* const* d_in, const int* in_sizes, int n_in,
                              void* d_out, int out_size, void* d_ws, size_t ws_size,
                              hipStream_t stream) {
    (void)in_sizes; (void)n_in; (void)out_size; (void)ws_size;

    const float* x         = (const float*)d_in[0];
    const float* ln_g      = (const float*)d_in[1];
    const float* ln_b      = (const float*)d_in[2];
    const float* in_proj_w = (const float*)d_in[3];
    const float* conv_w    = (const float*)d_in[4];
    const float* conv_b    = (const float*)d_in[5];
    const float* x_proj_w  = (const float*)d_in[6];
    const float* dt_proj_w = (const float*)d_in[7];
    const float* dt_proj_b = (const float*)d_in[8];
    const float* A_log     = (const float*)d_in[9];
    const float* Dp        = (const float*)d_in[10];
    const float* out_proj_w= (const float*)d_in[11];
    const float* outp_w    = (const float*)d_in[12];
    const float* outp_b    = (const float*)d_in[13];
    float* out = (float*)d_out;

    float* ws   = (float*)d_ws;
    float* hn   = ws;                                 // 4096*1024  (reused as mamba_out)
    float* xz   = hn   + (size_t)ROWS * DMODEL;       // 4096*4096
    float* ucv  = xz   + (size_t)ROWS * 2 * DINNER;   // 4096*2048
    float* xdbl = ucv  + (size_t)ROWS * DINNER;       // 4096*96
    float* dtb  = xdbl + (size_t)ROWS * XDBL_W;       // 4096*2048

    // 1) layernorm
    ln_kernel<<<ROWS, 256, 0, stream>>>(x, ln_g, ln_b, hn);

    // 2) in_proj: xz = hn @ in_proj_w^T   (4096 x 4096 x 1024)
    gemm_wmma_big_kernel<<<dim3(2 * DINNER / 64, ROWS / 128), 256, 0, stream>>>(
        hn, DMODEL, in_proj_w, DMODEL, xz, 2 * DINNER,
        ROWS, 2 * DINNER, DMODEL, nullptr, nullptr, 0);

    // 3) conv + SiLU
    conv_silu_kernel<<<(ROWS * DINNER) / 256, 256, 0, stream>>>(xz, conv_w, conv_b, ucv);

    // 4) x_proj: xdbl = ucv @ x_proj_w^T   (4096 x 96 x 2048) — small kernel
    gemm_wmma_kernel<<<dim3(XDBL_W / 32, ROWS / 64), 256, 0, stream>>>(
        ucv, DINNER, x_proj_w, DINNER, xdbl, XDBL_W,
        ROWS, XDBL_W, DINNER, nullptr, nullptr, 0);

    // 5) dt = softplus(dt_r @ dt_proj_w^T + b)   (4096 x 2048 x 64, lda=96)
    gemm_wmma_big_kernel<<<dim3(DINNER / 64, ROWS / 128), 256, 0, stream>>>(
        xdbl, XDBL_W, dt_proj_w, DTRANK, dtb, DINNER,
        ROWS, DINNER, DTRANK, dt_proj_b, nullptr, 1);

    // 6) selective scan + D skip + silu(z) gating (writes y into ucv)
    scan_kernel<<<(BATCH * DINNER) / 256, 256, 0, stream>>>(
        dtb, ucv, xz, xdbl, A_log, Dp);

    // 7) out_proj: hn = ucv @ out_proj_w^T   (4096 x 1024 x 2048)
    gemm_wmma_big_kernel<<<dim3(DMODEL / 64, ROWS / 128), 256, 0, stream>>>(
        ucv, DINNER, out_proj_w, DINNER, hn, DMODEL,
        ROWS, DMODEL, DINNER, nullptr, nullptr, 0);

    // 8) outp + bias + residual
    gemm_wmma_big_kernel<<<dim3(DMODEL / 64, ROWS / 128), 256, 0, stream>>>(
        hn, DMODEL, outp_w, DMODEL, out, DMODEL,
        ROWS, DMODEL, DMODEL, outp_b, x, 2);
}